// Megnet_GlobalModel_62689342653099
// MI455X (gfx1250) — compile-verified
//
#include <hip/hip_runtime.h>
#include <hip/hip_bf16.h>

typedef float v2f __attribute__((ext_vector_type(2)));
typedef float v8f __attribute__((ext_vector_type(8)));

#define DFEAT 64
#define BROWS 256
#define LDPAD 68   // LDS row pitch (floats) to break bank conflicts on column walks

// ---------------------------------------------------------------- zero scratch
__global__ void k_zero(float* __restrict__ p, int n) {
  int i = blockIdx.x * blockDim.x + threadIdx.x;
  if (i < n) p[i] = 0.0f;
}

// ------------------------------------------- edge_attr -> per-node sum + count
// 16 threads per edge, each handles 4 contiguous floats (float4 -> b128 load).
__global__ void k_edge_scatter(const float* __restrict__ edge_attr,
                               const long long* __restrict__ edge_src,
                               float* __restrict__ sum_e,
                               float* __restrict__ cnt_e, int E) {
  int gid = blockIdx.x * blockDim.x + threadIdx.x;
  if (gid >= E * 16) return;
  int e = gid >> 4;
  int q = gid & 15;
  int node = (int)edge_src[e];
  float4 v = ((const float4*)edge_attr)[(size_t)e * 16 + q];
  float* dst = sum_e + (size_t)node * DFEAT + q * 4;
  atomicAdd(dst + 0, v.x);
  atomicAdd(dst + 1, v.y);
  atomicAdd(dst + 2, v.z);
  atomicAdd(dst + 3, v.w);
  if (q == 0) atomicAdd(cnt_e + node, 1.0f);
}

// ------------- node level: u_e = sum_e/cnt -> scatter by batch; also scatter x
__global__ void k_node_scatter(const float* __restrict__ x,
                               const float* __restrict__ sum_e,
                               const float* __restrict__ cnt_e,
                               const long long* __restrict__ batch,
                               float* __restrict__ sum_ue,
                               float* __restrict__ sum_x,
                               float* __restrict__ cnt_b, int N) {
  int gid = blockIdx.x * blockDim.x + threadIdx.x;
  if (gid >= N * 16) return;
  int n = gid >> 4;
  int q = gid & 15;
  int b = (int)batch[n];
  float c = cnt_e[n];
  float inv = 1.0f / (c > 1.0f ? c : 1.0f);
  float4 se = ((const float4*)sum_e)[(size_t)n * 16 + q];
  float4 xv = ((const float4*)x)[(size_t)n * 16 + q];
  float* du = sum_ue + (size_t)b * DFEAT + q * 4;
  float* dx = sum_x + (size_t)b * DFEAT + q * 4;
  atomicAdd(du + 0, se.x * inv);
  atomicAdd(du + 1, se.y * inv);
  atomicAdd(du + 2, se.z * inv);
  atomicAdd(du + 3, se.w * inv);
  atomicAdd(dx + 0, xv.x);
  atomicAdd(dx + 1, xv.y);
  atomicAdd(dx + 2, xv.z);
  atomicAdd(dx + 3, xv.w);
  if (q == 0) atomicAdd(cnt_b + b, 1.0f);
}

// --------------------------------- comb[b] = [sum_ue/cb | sum_x/cb | u], [B,192]
__global__ void k_comb(const float* __restrict__ sum_ue,
                       const float* __restrict__ sum_x,
                       const float* __restrict__ cnt_b,
                       const float* __restrict__ u,
                       float* __restrict__ comb, int B) {
  int gid = blockIdx.x * blockDim.x + threadIdx.x;
  if (gid >= B * 3 * DFEAT) return;
  int b = gid / (3 * DFEAT);
  int c = gid % (3 * DFEAT);
  float cb = cnt_b[b];
  float inv = 1.0f / (cb > 1.0f ? cb : 1.0f);
  float v;
  if (c < DFEAT)           v = sum_ue[(size_t)b * DFEAT + c] * inv;
  else if (c < 2 * DFEAT)  v = sum_x[(size_t)b * DFEAT + (c - DFEAT)] * inv;
  else                     v = u[(size_t)b * DFEAT + (c - 2 * DFEAT)];
  comb[gid] = v;
}

// ------------------- one Linear(+bias) -> ReLU -> BatchNorm layer, WMMA f32 x4
// in:  [256, K]   (pitch ldin), global or LDS (generic pointer)
// W:   [64, K]    row-major (out-feature major), B-matrix is W^T
// out: [256, 64]  in LDS with pitch ldout
__device__ __forceinline__ void mlp_layer(
    const float* __restrict__ in, int ldin, int K,
    const float* __restrict__ W, const float* __restrict__ bias,
    const float* __restrict__ gamma, const float* __restrict__ beta,
    float* out, int ldout, float* s_scale, float* s_shift, int tid) {
  const int wave  = tid >> 5;
  const int lane  = tid & 31;
  const int m0    = lane & 15;
  const int half2 = (lane >> 4) * 2;  // K offset within x4 step (ISA A/B layout)
  const int half8 = (lane >> 4) * 8;  // M offset within C/D tile (ISA C layout)

  // 64 output tiles (16 M-tiles x 4 N-tiles), 2 per wave
  for (int t = wave; t < 64; t += 32) {
    const int mt = (t >> 2) << 4;
    const int nt = (t & 3) << 4;
    const int ncol = nt + m0;
    v8f c;
    const float bv = bias[ncol];
#pragma unroll
    for (int r = 0; r < 8; ++r) c[r] = bv;

    const float* arow = in + (size_t)(mt + m0) * ldin;
    const float* bcol = W + (size_t)ncol * K;  // column ncol of W^T
    for (int k0 = 0; k0 < K; k0 += 4) {
      v2f a, b;
      a[0] = arow[k0 + half2];
      a[1] = arow[k0 + half2 + 1];
      b[0] = bcol[k0 + half2];
      b[1] = bcol[k0 + half2 + 1];
      c = __builtin_amdgcn_wmma_f32_16x16x4_f32(
          /*neg_a=*/false, a, /*neg_b=*/false, b,
          /*c_mod=*/(short)0, c, /*reuse_a=*/false, /*reuse_b=*/false);
    }
#pragma unroll
    for (int r = 0; r < 8; ++r) {
      float v = c[r];
      v = v > 0.0f ? v : 0.0f;  // ReLU
      out[(size_t)(mt + half8 + r) * ldout + ncol] = v;
    }
  }
  __syncthreads();

  // BatchNorm batch stats (biased variance) per column over 256 rows
  if (tid < DFEAT) {
    float s = 0.0f, ss = 0.0f;
    for (int r = 0; r < BROWS; ++r) {
      float v = out[(size_t)r * ldout + tid];
      s += v;
      ss += v * v;
    }
    const float mu = s * (1.0f / BROWS);
    float var = ss * (1.0f / BROWS) - mu * mu;
    var = var < 0.0f ? 0.0f : var;
    const float sc = gamma[tid] * rsqrtf(var + 1e-5f);
    s_scale[tid] = sc;
    s_shift[tid] = beta[tid] - mu * sc;
  }
  __syncthreads();

  for (int i = tid; i < BROWS * DFEAT; i += 1024) {
    const int r = i >> 6, cc = i & 63;
    out[(size_t)r * ldout + cc] =
        out[(size_t)r * ldout + cc] * s_scale[cc] + s_shift[cc];
  }
  __syncthreads();
}

__global__ __launch_bounds__(1024) void k_mlp(
    const float* __restrict__ comb,
    const float* __restrict__ W0, const float* __restrict__ b0,
    const float* __restrict__ W1, const float* __restrict__ b1,
    const float* __restrict__ W2, const float* __restrict__ b2,
    const float* __restrict__ g0, const float* __restrict__ be0,
    const float* __restrict__ g1, const float* __restrict__ be1,
    const float* __restrict__ g2, const float* __restrict__ be2,
    float* __restrict__ outg) {
  extern __shared__ float smem[];
  float* bufA   = smem;                       // [256][68]
  float* bufB   = smem + BROWS * LDPAD;       // [256][68]
  float* sScale = smem + 2 * BROWS * LDPAD;   // [64]
  float* sShift = sScale + DFEAT;             // [64]
  const int tid = threadIdx.x;

  mlp_layer(comb, 3 * DFEAT, 3 * DFEAT, W0, b0, g0, be0, bufA, LDPAD, sScale, sShift, tid);
  mlp_layer(bufA, LDPAD, DFEAT, W1, b1, g1, be1, bufB, LDPAD, sScale, sShift, tid);
  mlp_layer(bufB, LDPAD, DFEAT, W2, b2, g2, be2, bufA, LDPAD, sScale, sShift, tid);

  for (int i = tid; i < BROWS * DFEAT; i += 1024)
    outg[i] = bufA[(size_t)(i >> 6) * LDPAD + (i & 63)];
}

extern "C" void kernel_launch(void* const* d_in, const int* in_sizes, int n_in,
                              void* d_out, int out_size, void* d_ws, size_t ws_size,
                              hipStream_t stream) {
  const float*     x         = (const float*)d_in[0];
  const float*     edge_attr = (const float*)d_in[1];
  const float*     u         = (const float*)d_in[2];
  const long long* edge_idx  = (const long long*)d_in[3];
  const long long* batch     = (const long long*)d_in[4];
  const float *W0 = (const float*)d_in[5],  *b0 = (const float*)d_in[6];
  const float *W1 = (const float*)d_in[7],  *b1 = (const float*)d_in[8];
  const float *W2 = (const float*)d_in[9],  *b2 = (const float*)d_in[10];
  const float *g0 = (const float*)d_in[11], *be0 = (const float*)d_in[12];
  const float *g1 = (const float*)d_in[13], *be1 = (const float*)d_in[14];
  const float *g2 = (const float*)d_in[15], *be2 = (const float*)d_in[16];
  float* out = (float*)d_out;

  const int N = in_sizes[0] / DFEAT;   // 50000
  const int E = in_sizes[1] / DFEAT;   // 800000
  const int B = in_sizes[2] / DFEAT;   // 256

  // workspace layout (floats)
  float* wsf = (float*)d_ws;
  size_t off = 0;
  float* sum_e  = wsf + off; off += (size_t)N * DFEAT;
  float* cnt_e  = wsf + off; off += (size_t)N;
  float* sum_ue = wsf + off; off += (size_t)B * DFEAT;
  float* sum_x  = wsf + off; off += (size_t)B * DFEAT;
  float* cnt_b  = wsf + off; off += (size_t)B;
  float* comb   = wsf + off; off += (size_t)B * 3 * DFEAT;
  const int nzero = (int)off;  // ~3.33M floats, must be re-zeroed every call

  k_zero<<<(nzero + 255) / 256, 256, 0, stream>>>(wsf, nzero);

  int t1 = E * 16;
  k_edge_scatter<<<(t1 + 255) / 256, 256, 0, stream>>>(edge_attr, edge_idx,
                                                       sum_e, cnt_e, E);
  int t2 = N * 16;
  k_node_scatter<<<(t2 + 255) / 256, 256, 0, stream>>>(x, sum_e, cnt_e, batch,
                                                       sum_ue, sum_x, cnt_b, N);
  int t3 = B * 3 * DFEAT;
  k_comb<<<(t3 + 255) / 256, 256, 0, stream>>>(sum_ue, sum_x, cnt_b, u, comb, B);

  size_t shmem = (size_t)(2 * BROWS * LDPAD + 2 * DFEAT) * sizeof(float);
  k_mlp<<<1, 1024, shmem, stream>>>(comb, W0, b0, W1, b1, W2, b2,
                                    g0, be0, g1, be1, g2, be2, out);
}